// LocalGlobalAdaptiveFusionV2_18631568130347
// MI455X (gfx1250) — compile-verified
//
#include <hip/hip_runtime.h>
#include <math.h>

#define N_GRAPHS 512
#define HIDDEN   256
#define LN_EPS   1e-5f
#define COS_EPS  1e-8f

typedef __attribute__((ext_vector_type(2))) float v2f;
typedef __attribute__((ext_vector_type(8))) float v8f;

// d_ws layout (floats):
//  [0,512)      n per graph
//  [512,1024)   e (intra-edge count) per graph
//  [1024,1536)  divergence per graph
//  [1536,2560)  w (512 x 2) softmax gate weights

__global__ void k_zero(float* __restrict__ e) {
    int t = threadIdx.x;
    if (t < N_GRAPHS) e[t] = 0.0f;
}

// One block per graph: binary-search node range in sorted batch, then
// column-parallel mean pooling of both feature arrays + cosine similarity.
__global__ void k_pool(const float* __restrict__ xg, const float* __restrict__ xa,
                       const int* __restrict__ batch, int n_nodes,
                       float* __restrict__ nOut, float* __restrict__ divOut) {
    __shared__ float sdot[256], sg2[256], sa2[256];
    __shared__ int sRange[2];
    const int g = blockIdx.x;
    const int t = threadIdx.x;
    if (t == 0) {
        int lo = 0, hi = n_nodes;
        while (lo < hi) { int mid = (lo + hi) >> 1; if (batch[mid] < g) lo = mid + 1; else hi = mid; }
        sRange[0] = lo;
        hi = n_nodes;
        while (lo < hi) { int mid = (lo + hi) >> 1; if (batch[mid] < g + 1) lo = mid + 1; else hi = mid; }
        sRange[1] = lo;
    }
    __syncthreads();
    const int start = sRange[0], end = sRange[1];
    const float n = (float)(end - start);
    float sgc = 0.0f, sac = 0.0f;
    for (int i = start; i < end; ++i) {
        sgc += xg[(size_t)i * HIDDEN + t];
        sac += xa[(size_t)i * HIDDEN + t];
    }
    const float pg = sgc / n;
    const float pa = sac / n;
    sdot[t] = pg * pa; sg2[t] = pg * pg; sa2[t] = pa * pa;
    __syncthreads();
    for (int s = 128; s > 0; s >>= 1) {
        if (t < s) { sdot[t] += sdot[t + s]; sg2[t] += sg2[t + s]; sa2[t] += sa2[t + s]; }
        __syncthreads();
    }
    if (t == 0) {
        float ng = sqrtf(sg2[0]), na = sqrtf(sa2[0]);
        float cosv = sdot[0] / (fmaxf(ng, COS_EPS) * fmaxf(na, COS_EPS));
        nOut[g]   = n;
        divOut[g] = (1.0f - cosv) * 0.5f;
    }
}

// Intra-graph edge counts. ~0.2% of edges hit, so direct atomics are cheap.
// All addends are 1.0f -> order-independent exact integer sums (< 2^24).
__global__ void k_edges(const int* __restrict__ ei, const int* __restrict__ batch,
                        int n_edges, float* __restrict__ eOut) {
    const int stride = gridDim.x * blockDim.x;
    for (int idx = blockIdx.x * blockDim.x + threadIdx.x; idx < n_edges; idx += stride) {
        int s = ei[idx];
        int d = ei[n_edges + idx];
        int gs = batch[s];
        int gd = batch[d];
        if (gs == gd) atomicAdd(&eOut[gs], 1.0f);
    }
}

// Gate MLP. One wave per 16 graphs. First GEMM (feats[16x4] @ W1[4x32]) uses
// V_WMMA_F32_16X16X4_F32 (two N-tiles). LayerNorm+rest staged through LDS.
__global__ void __launch_bounds__(32) k_mlp(
    const float* __restrict__ W1, const float* __restrict__ b1,
    const float* __restrict__ ln_g, const float* __restrict__ ln_b,
    const float* __restrict__ W2, const float* __restrict__ b2,
    const float* __restrict__ W3, const float* __restrict__ b3,
    const float* __restrict__ nArr, const float* __restrict__ eArr,
    const float* __restrict__ divArr, float* __restrict__ wOut) {
    __shared__ float feats[16][4];
    __shared__ float h1[16][32];
    const int lane = threadIdx.x;
    const int g0 = blockIdx.x * 16;

    if (lane < 16) {
        int g = g0 + lane;
        float n = nArr[g];
        float e = eArr[g];
        feats[lane][0] = logf(n + 1.0f) * (1.0f / logf(501.0f));           // scale
        feats[lane][1] = e / (n * (n - 1.0f) + 1e-8f);                     // density
        feats[lane][2] = fminf(e / (n + 1e-8f) * 0.1f, 1.0f);              // avg_deg
        feats[lane][3] = divArr[g];                                        // divergence
    }
    __syncthreads();

    // A-frag (16x4 f32): lanes 0-15 hold M=lane, K=0/1; lanes 16-31 hold K=2/3.
    const int half = lane >> 4;
    const int m    = lane & 15;
    v2f a;
    a.x = feats[m][2 * half + 0];
    a.y = feats[m][2 * half + 1];
    // B-frag (4x16 f32), mirrored layout: lanes hold N, VGPR+lane-half hold K.
    // W1 is [4][32] row-major.
    v2f bB0, bB1;
    bB0.x = W1[(2 * half + 0) * 32 + m];
    bB0.y = W1[(2 * half + 1) * 32 + m];
    bB1.x = W1[(2 * half + 0) * 32 + 16 + m];
    bB1.y = W1[(2 * half + 1) * 32 + 16 + m];

    v8f c0 = {};
    v8f c1 = {};
    c0 = __builtin_amdgcn_wmma_f32_16x16x4_f32(false, a, false, bB0, (short)0, c0, false, false);
    c1 = __builtin_amdgcn_wmma_f32_16x16x4_f32(false, a, false, bB1, (short)0, c1, false, false);

    // C/D layout: VGPR v, lanes 0-15 -> (M=v, N=lane); lanes 16-31 -> (M=v+8, N=lane-16).
    union { v8f v; float f[8]; } u0, u1;
    u0.v = c0; u1.v = c1;
#pragma unroll
    for (int v = 0; v < 8; ++v) {
        int M = v + 8 * half;
        h1[M][m]      = u0.f[v] + b1[m];
        h1[M][16 + m] = u1.f[v] + b1[16 + m];
    }
    __syncthreads();

    if (lane < 16) {
        float hr[32];
        float mean = 0.0f;
#pragma unroll
        for (int j = 0; j < 32; ++j) { hr[j] = h1[lane][j]; mean += hr[j]; }
        mean *= (1.0f / 32.0f);
        float var = 0.0f;
#pragma unroll
        for (int j = 0; j < 32; ++j) { float d = hr[j] - mean; var += d * d; }
        var *= (1.0f / 32.0f);
        float inv = rsqrtf(var + LN_EPS);
#pragma unroll
        for (int j = 0; j < 32; ++j)
            hr[j] = fmaxf((hr[j] - mean) * inv * ln_g[j] + ln_b[j], 0.0f);

        float h2[16];
#pragma unroll
        for (int j = 0; j < 16; ++j) {
            float acc = b2[j];
#pragma unroll
            for (int k = 0; k < 32; ++k) acc += hr[k] * W2[k * 16 + j];
            h2[j] = fmaxf(acc, 0.0f);
        }
        float r0 = b3[0], r1 = b3[1];
#pragma unroll
        for (int j = 0; j < 16; ++j) { r0 += h2[j] * W3[j * 2 + 0]; r1 += h2[j] * W3[j * 2 + 1]; }
        r0 *= 2.0f; r1 *= 2.0f;                       // raw / TEMP, TEMP = 0.5
        float mx = fmaxf(r0, r1);
        float e0 = expf(r0 - mx), e1 = expf(r1 - mx);
        float inv_s = 1.0f / (e0 + e1);
        int g = g0 + lane;
        wOut[2 * g + 0] = e0 * inv_s;
        wOut[2 * g + 1] = e1 * inv_s;
    }
}

// Final blend, float4-vectorized: out = w0*x_ggnn + w1*x_appnp.
__global__ void k_combine(const float4* __restrict__ xg, const float4* __restrict__ xa,
                          const int* __restrict__ batch, const float* __restrict__ w,
                          float4* __restrict__ out, int total4) {
    int idx = blockIdx.x * blockDim.x + threadIdx.x;
    if (idx >= total4) return;
    int node = idx >> 6;                 // HIDDEN/4 == 64 float4 per node
    int g = batch[node];
    float w0 = w[2 * g + 0];
    float w1 = w[2 * g + 1];
    float4 A = xg[idx];
    float4 B = xa[idx];
    float4 o;
    o.x = w0 * A.x + w1 * B.x;
    o.y = w0 * A.y + w1 * B.y;
    o.z = w0 * A.z + w1 * B.z;
    o.w = w0 * A.w + w1 * B.w;
    out[idx] = o;
}

extern "C" void kernel_launch(void* const* d_in, const int* in_sizes, int n_in,
                              void* d_out, int out_size, void* d_ws, size_t ws_size,
                              hipStream_t stream) {
    const float* xg   = (const float*)d_in[0];
    const float* xa   = (const float*)d_in[1];
    const float* W1   = (const float*)d_in[2];
    const float* b1   = (const float*)d_in[3];
    const float* ln_g = (const float*)d_in[4];
    const float* ln_b = (const float*)d_in[5];
    const float* W2   = (const float*)d_in[6];
    const float* b2   = (const float*)d_in[7];
    const float* W3   = (const float*)d_in[8];
    const float* b3   = (const float*)d_in[9];
    const int*   ei    = (const int*)d_in[10];
    const int*   batch = (const int*)d_in[11];

    const int n_nodes = in_sizes[11];
    const int n_edges = in_sizes[10] / 2;

    float* ws     = (float*)d_ws;
    float* nArr   = ws;
    float* eArr   = ws + 512;
    float* divArr = ws + 1024;
    float* wArr   = ws + 1536;

    k_zero<<<1, 512, 0, stream>>>(eArr);
    k_pool<<<N_GRAPHS, 256, 0, stream>>>(xg, xa, batch, n_nodes, nArr, divArr);
    k_edges<<<512, 256, 0, stream>>>(ei, batch, n_edges, eArr);
    k_mlp<<<N_GRAPHS / 16, 32, 0, stream>>>(W1, b1, ln_g, ln_b, W2, b2, W3, b3,
                                            nArr, eArr, divArr, wArr);
    const int total4 = n_nodes * (HIDDEN / 4);
    k_combine<<<(total4 + 255) / 256, 256, 0, stream>>>(
        (const float4*)xg, (const float4*)xa, batch, wArr, (float4*)d_out, total4);
}